// StandardPositionEmbedding_1030792151465
// MI455X (gfx1250) — compile-verified
//
#include <hip/hip_runtime.h>
#include <hip/hip_bf16.h>

// out[b,s,d] = x[b,s,d] + pe[s,d]
// pe[s,2k]   = sin(s * c_k),  pe[s,2k+1] = cos(s * c_k)
// c_k = exp(-2k*ln(1000)/2048) = exp2(k * (-2*log2(1000)/2048))
//
// Streaming, bandwidth-bound kernel for MI455X (gfx1250):
//  - float4 (B128) global loads/stores
//  - non-temporal cache hints (read-once / write-once traffic, keep L2 clean)
//  - PE computed once per (s, d-vec) and reused across the 8 batch slices
//    (8 outstanding B128 loads per thread for latency hiding)

typedef float __attribute__((ext_vector_type(4))) v4f;

__global__ __launch_bounds__(256) void pe_add_kernel(const float* __restrict__ x,
                                                     float* __restrict__ out) {
    constexpr int S = 4096;
    constexpr int D = 2048;
    constexpr int B = 8;
    constexpr int VECS_PER_ROW = D / 4;              // 512 float4 per row
    constexpr size_t SLICE = (size_t)S * D;          // elements per batch slice

    const int vid = blockIdx.x * blockDim.x + threadIdx.x;   // over S * D/4
    if (vid >= S * VECS_PER_ROW) return;

    const int s  = vid >> 9;           // vid / 512
    const int j  = vid & 511;          // float4 index within row
    const int d0 = j << 2;             // base dim of this float4 (covers pairs 2j, 2j+1)

    // c_k = exp2(k * C2), C2 = -2*log2(1000)/2048
    const float C2 = -0.009732211215490319f;
    const float k0 = (float)(d0 >> 1);               // = 2j
    const float c0 = __builtin_amdgcn_exp2f(k0 * C2);          // v_exp_f32
    const float c1 = __builtin_amdgcn_exp2f((k0 + 1.0f) * C2); // v_exp_f32

    const float sf = (float)s;
    const float a0 = sf * c0;
    const float a1 = sf * c1;

    v4f pe;
    pe.x = __sinf(a0);   // dim d0     (even -> sin)  v_sin_f32
    pe.y = __cosf(a0);   // dim d0 + 1 (odd  -> cos)  v_cos_f32
    pe.z = __sinf(a1);   // dim d0 + 2
    pe.w = __cosf(a1);   // dim d0 + 3

    const size_t off = (size_t)s * D + d0;

#pragma unroll
    for (int b = 0; b < B; ++b) {
        const v4f* __restrict__ src = (const v4f*)(x + (size_t)b * SLICE + off);
        v4f*       __restrict__ dst = (v4f*)(out + (size_t)b * SLICE + off);
        v4f v = __builtin_nontemporal_load(src);     // GLOBAL_LOAD_B128, TH=NT
        v += pe;
        __builtin_nontemporal_store(v, dst);         // GLOBAL_STORE_B128, TH=NT
    }
}

extern "C" void kernel_launch(void* const* d_in, const int* in_sizes, int n_in,
                              void* d_out, int out_size, void* d_ws, size_t ws_size,
                              hipStream_t stream) {
    const float* x = (const float*)d_in[0];
    float* out = (float*)d_out;

    constexpr int S = 4096, D = 2048;
    const int total_vecs = S * (D / 4);              // 2,097,152 threads
    const int block = 256;                            // 8 wave32 waves per block
    const int grid = (total_vecs + block - 1) / block; // 8192 blocks

    pe_add_kernel<<<grid, block, 0, stream>>>(x, out);
}